// WeightedLoss_21569325760611
// MI455X (gfx1250) — compile-verified
//
#include <hip/hip_runtime.h>

typedef __attribute__((ext_vector_type(2))) float v2f;
typedef __attribute__((ext_vector_type(8))) float v8f;

// argmax-first semantics matching jnp.argmax (lowest index wins ties)
__device__ __forceinline__ float sample_loss(float p0, float p1, float p2, float t,
                                             float w0, float w1, float w2) {
    float best = p0;
    float w    = w0;
    if (p1 > best) { best = p1; w = w1; }
    if (p2 > best) { best = p2; w = w2; }
    float d = best - t;
    return w * d * d;
}

// Kernel 1: grid-stride over quads of 4 samples (12 floats of pred = 3x b128 loads,
// 4 labels = 1x b128 load per thread-iteration). Deterministic block partial sums.
__global__ void wl_partial(const float* __restrict__ pred,
                           const float* __restrict__ labels,
                           const float* __restrict__ weights,
                           float* __restrict__ partial,
                           int nquads, int nsamples) {
    const float w0 = weights[0];
    const float w1 = weights[1];
    const float w2 = weights[2];

    const float4* __restrict__ p4 = reinterpret_cast<const float4*>(pred);
    const float4* __restrict__ t4 = reinterpret_cast<const float4*>(labels);

    float acc = 0.0f;
    const int stride = gridDim.x * blockDim.x;
    for (int q = blockIdx.x * blockDim.x + threadIdx.x; q < nquads; q += stride) {
        float4 a = p4[3 * q + 0];   // s0: a.x a.y a.z | s1: a.w ...
        float4 b = p4[3 * q + 1];   // s1: b.x b.y | s2: b.z b.w ...
        float4 c = p4[3 * q + 2];   // s2: c.x | s3: c.y c.z c.w
        float4 t = t4[q];
        acc += sample_loss(a.x, a.y, a.z, t.x, w0, w1, w2);
        acc += sample_loss(a.w, b.x, b.y, t.y, w0, w1, w2);
        acc += sample_loss(b.z, b.w, c.x, t.z, w0, w1, w2);
        acc += sample_loss(c.y, c.z, c.w, t.w, w0, w1, w2);
    }

    // Tail (B not divisible by 4) — handled by one thread; B=8M is divisible, so no-op here.
    if (blockIdx.x == 0 && threadIdx.x == 0) {
        for (int i = nquads * 4; i < nsamples; ++i) {
            acc += sample_loss(pred[3 * i + 0], pred[3 * i + 1], pred[3 * i + 2],
                               labels[i], w0, w1, w2);
        }
    }

    // wave32 tree reduction
    #pragma unroll
    for (int m = 1; m < 32; m <<= 1)
        acc += __shfl_xor(acc, m, 32);

    __shared__ float smem[8];   // 256 threads -> 8 wave32
    const int lane = threadIdx.x & 31;
    const int wave = threadIdx.x >> 5;
    if (lane == 0) smem[wave] = acc;
    __syncthreads();
    if (threadIdx.x == 0) {
        float r = 0.0f;
        const int nw = blockDim.x >> 5;
        for (int i = 0; i < nw; ++i) r += smem[i];
        partial[blockIdx.x] = r;   // deterministic per-block partial
    }
}

// Kernel 2: single wave32 reduces npart partials; the 32 lane accumulators are
// collapsed with one V_WMMA_F32_16X16X4_F32 (A = lane values, B = ones):
//   D[M][N] = v_M + v_{M+16}  (constant across N)
// Lane n in [0,16) holds D[0..7][n] in its 8 D-VGPRs  -> sum = v_0..7 + v_16..23
// Lane n in [16,32) holds D[8..15][n-16]              -> sum = v_8..15 + v_24..31
// One shfl_xor(16) completes the scalar.
__global__ void wl_final(const float* __restrict__ partial, int npart,
                         float* __restrict__ out) {
    float v = 0.0f;
    for (int i = threadIdx.x; i < npart; i += 32)
        v += partial[i];

    v2f a; a[0] = v;    a[1] = 0.0f;   // A 16x4: K0/K2 = lane values, K1/K3 = 0
    v2f b; b[0] = 1.0f; b[1] = 1.0f;   // B 4x16: all ones
    v8f c = {};
    v8f d = __builtin_amdgcn_wmma_f32_16x16x4_f32(
        /*neg_a=*/false, a, /*neg_b=*/false, b,
        /*c_mod=*/(short)0, c, /*reuse_a=*/false, /*reuse_b=*/false);

    float t = ((d[0] + d[1]) + (d[2] + d[3])) + ((d[4] + d[5]) + (d[6] + d[7]));
    float tot = t + __shfl_xor(t, 16, 32);
    if (threadIdx.x == 0) out[0] = tot;
}

extern "C" void kernel_launch(void* const* d_in, const int* in_sizes, int n_in,
                              void* d_out, int out_size, void* d_ws, size_t ws_size,
                              hipStream_t stream) {
    const float* pred    = (const float*)d_in[0];  // [B,3] fp32
    const float* labels  = (const float*)d_in[1];  // [B]   fp32
    const float* weights = (const float*)d_in[2];  // [3]   fp32
    float* out = (float*)d_out;

    const int B      = in_sizes[1];
    const int nquads = B >> 2;

    int nblk = 2048;
    if (ws_size < (size_t)nblk * sizeof(float)) {
        nblk = (int)(ws_size / sizeof(float));
        if (nblk < 1) nblk = 1;
    }
    float* partial = (float*)d_ws;

    wl_partial<<<nblk, 256, 0, stream>>>(pred, labels, weights, partial, nquads, B);
    wl_final<<<1, 32, 0, stream>>>(partial, nblk, out);
}